// Attention_5583457485032
// MI455X (gfx1250) — compile-verified
//
#include <hip/hip_runtime.h>
#include <math.h>

typedef __attribute__((ext_vector_type(2))) float v2f;
typedef __attribute__((ext_vector_type(8))) float v8f;

#define N_NODES 200000
#define EMB 256
#define CS_BLOCKS 1024
#define PASS_BLOCKS 512
#define WAVES_PER_BLOCK 8
#define EPS 1e-12f

__device__ __forceinline__ float wave_sum32(float v) {
    v += __shfl_xor(v, 1, 32);
    v += __shfl_xor(v, 2, 32);
    v += __shfl_xor(v, 4, 32);
    v += __shfl_xor(v, 8, 32);
    v += __shfl_xor(v, 16, 32);
    return v;
}

// ---- Pass 1: per-block partial column sums of x. float4 (b128) streaming,
// 4 row-substreams per block, deterministic LDS combine (no atomics).
__global__ __launch_bounds__(256) void colsum_partial(const float* __restrict__ x,
                                                      float* __restrict__ part) {
    __shared__ float4 sdata[256];
    const int t   = threadIdx.x;
    const int c4  = t & 63;   // float4 column group (covers cols 4*c4 .. 4*c4+3)
    const int sub = t >> 6;   // row substream 0..3
    const float4* X4 = (const float4*)x;

    float4 acc = make_float4(0.f, 0.f, 0.f, 0.f);
    for (int r = blockIdx.x * 4 + sub; r < N_NODES; r += CS_BLOCKS * 4) {
        const float4 v = X4[(size_t)r * 64 + c4];
        acc.x += v.x; acc.y += v.y; acc.z += v.z; acc.w += v.w;
    }
    sdata[t] = acc;
    __syncthreads();
    if (t < 64) {
        const float4 s0 = sdata[t], s1 = sdata[64 + t], s2 = sdata[128 + t], s3 = sdata[192 + t];
        part[blockIdx.x * EMB + 4 * t + 0] = ((s0.x + s1.x) + s2.x) + s3.x;
        part[blockIdx.x * EMB + 4 * t + 1] = ((s0.y + s1.y) + s2.y) + s3.y;
        part[blockIdx.x * EMB + 4 * t + 2] = ((s0.z + s1.z) + s2.z) + s3.z;
        part[blockIdx.x * EMB + 4 * t + 3] = ((s0.w + s1.w) + s2.w) + s3.w;
    }
}

__global__ __launch_bounds__(256) void reduce_colsum(const float* __restrict__ part,
                                                     float* __restrict__ temp) {
    const int t = threadIdx.x;
    float s = 0.0f;
    for (int b = 0; b < CS_BLOCKS; ++b) s += part[b * EMB + t];
    temp[t] = s * (1.0f / (float)N_NODES);
}

// ---- Projection: h[head] = tanh(temp_row @ W[head]) via V_WMMA_F32_16X16X4_F32.
// grid = 32 blocks (head = blk>>4, nTile = blk&15), block = 1 wave.
// A (16x4 f32): lane L<16 holds {A[L][0],A[L][1]}, lane L>=16 holds {A[L-16][2],A[L-16][3]};
// only row M=0 (temp) is nonzero -> built branch-free with v_cndmask (no exec divergence).
// B (4x16): lane n<16 holds {B[0][n],B[1][n]}, lane n>=16 holds {B[2][n-16],B[3][n-16]}.
// Two accumulators (even/odd k) halve the WMMA dependency chain.
__global__ __launch_bounds__(32) void proj_wmma(const float* __restrict__ temp, int tempStride,
                                                const float* __restrict__ W,
                                                float* __restrict__ h) {
    const int head = blockIdx.x >> 4;
    const int nT   = blockIdx.x & 15;
    const int lane = threadIdx.x;
    const float* tv = temp + head * tempStride;
    const float* Wi = W + (size_t)head * EMB * EMB;
    const int col   = nT * 16 + (lane & 15);
    const int koff  = (lane < 16) ? 0 : 2;
    const bool keep = ((lane & 15) == 0);   // lanes 0 and 16 carry the temp row

    v8f c0 = {}, c1 = {};
    for (int k = 0; k < 64; k += 2) {
        {
            const float vx = tv[4 * k + koff];
            const float vy = tv[4 * k + koff + 1];
            v2f a; a.x = keep ? vx : 0.0f; a.y = keep ? vy : 0.0f;
            v2f b;
            b.x = Wi[(size_t)(4 * k + koff)     * EMB + col];
            b.y = Wi[(size_t)(4 * k + koff + 1) * EMB + col];
            c0 = __builtin_amdgcn_wmma_f32_16x16x4_f32(false, a, false, b,
                                                       (short)0, c0, false, false);
        }
        {
            const int k1 = k + 1;
            const float vx = tv[4 * k1 + koff];
            const float vy = tv[4 * k1 + koff + 1];
            v2f a; a.x = keep ? vx : 0.0f; a.y = keep ? vy : 0.0f;
            v2f b;
            b.x = Wi[(size_t)(4 * k1 + koff)     * EMB + col];
            b.y = Wi[(size_t)(4 * k1 + koff + 1) * EMB + col];
            c1 = __builtin_amdgcn_wmma_f32_16x16x4_f32(false, a, false, b,
                                                       (short)0, c1, false, false);
        }
    }
    if (lane < 16) h[head * EMB + nT * 16 + lane] = tanhf(c0[0] + c1[0]);
}

// ---- Streaming attention pass. One wave per row (grid-strided), float4 loads.
// SECOND==0: score = x_r . h_i ;  att1 = sigmoid(score/max(|score|,eps)); out1 += att1 * x_r
// SECOND==1: score = att1 * (x_r . h_i); att2 = sigmoid(...); out2 += att2*att1 * x_r
template <int SECOND>
__global__ __launch_bounds__(256) void pass_attn(const float* __restrict__ x,
                                                 const float* __restrict__ h,        // [2][256]
                                                 const float* __restrict__ att_prev, // [2][N] (SECOND only)
                                                 float* __restrict__ att_out,        // [2][N] (first only)
                                                 float* __restrict__ part) {         // [PASS_BLOCKS][512]
    __shared__ float sh[512];
    __shared__ float wacc[WAVES_PER_BLOCK * 512];
    const int t = threadIdx.x;
    sh[t]       = h[t];
    sh[t + 256] = h[t + 256];
    __syncthreads();

    const int w = t >> 5, lane = t & 31;
    const int gw = blockIdx.x * WAVES_PER_BLOCK + w;
    const int GW = PASS_BLOCKS * WAVES_PER_BLOCK;
    const float4* X4 = (const float4*)x;

    float4 acc[2][2];
#pragma unroll
    for (int i = 0; i < 2; ++i)
#pragma unroll
        for (int p = 0; p < 2; ++p) acc[i][p] = make_float4(0.f, 0.f, 0.f, 0.f);

    for (int r = gw; r < N_NODES; r += GW) {
        const float4 a0 = X4[(size_t)r * 64 + lane];
        const float4 a1 = X4[(size_t)r * 64 + 32 + lane];
        float ap[2] = {1.0f, 1.0f};
        if (SECOND) {
            ap[0] = att_prev[r];
            ap[1] = att_prev[N_NODES + r];
        }
#pragma unroll
        for (int i = 0; i < 2; ++i) {
            const float* hv = sh + i * 256;
            float d = a0.x * hv[4 * lane + 0] + a0.y * hv[4 * lane + 1] +
                      a0.z * hv[4 * lane + 2] + a0.w * hv[4 * lane + 3] +
                      a1.x * hv[128 + 4 * lane + 0] + a1.y * hv[128 + 4 * lane + 1] +
                      a1.z * hv[128 + 4 * lane + 2] + a1.w * hv[128 + 4 * lane + 3];
            d = wave_sum32(d);
            const float score = SECOND ? (ap[i] * d) : d;
            const float nrm = fmaxf(fabsf(score), EPS);
            const float att = 1.0f / (1.0f + expf(-(score / nrm)));
            if (!SECOND && lane == 0) att_out[(size_t)i * N_NODES + r] = att;
            const float wgt = SECOND ? (att * ap[i]) : att;
            acc[i][0].x += wgt * a0.x; acc[i][0].y += wgt * a0.y;
            acc[i][0].z += wgt * a0.z; acc[i][0].w += wgt * a0.w;
            acc[i][1].x += wgt * a1.x; acc[i][1].y += wgt * a1.y;
            acc[i][1].z += wgt * a1.z; acc[i][1].w += wgt * a1.w;
        }
    }

    // Deterministic per-block reduction: wave accumulators -> LDS -> fixed-order sum.
#pragma unroll
    for (int i = 0; i < 2; ++i) {
        float* dst = wacc + w * 512 + i * 256;
        dst[4 * lane + 0]       = acc[i][0].x;
        dst[4 * lane + 1]       = acc[i][0].y;
        dst[4 * lane + 2]       = acc[i][0].z;
        dst[4 * lane + 3]       = acc[i][0].w;
        dst[128 + 4 * lane + 0] = acc[i][1].x;
        dst[128 + 4 * lane + 1] = acc[i][1].y;
        dst[128 + 4 * lane + 2] = acc[i][1].z;
        dst[128 + 4 * lane + 3] = acc[i][1].w;
    }
    __syncthreads();
    float s0 = 0.0f, s1 = 0.0f;
    for (int wv = 0; wv < WAVES_PER_BLOCK; ++wv) {
        s0 += wacc[wv * 512 + t];
        s1 += wacc[wv * 512 + 256 + t];
    }
    part[blockIdx.x * 512 + t]       = s0;
    part[blockIdx.x * 512 + 256 + t] = s1;
}

__global__ __launch_bounds__(256) void reduce_part_scale(const float* __restrict__ part, int nb,
                                                         float* __restrict__ out, float scale) {
    const int t = threadIdx.x;
    float s0 = 0.0f, s1 = 0.0f;
    for (int b = 0; b < nb; ++b) {
        s0 += part[b * 512 + t];
        s1 += part[b * 512 + 256 + t];
    }
    out[t]       = s0 * scale;
    out[t + 256] = s1 * scale;
}

extern "C" void kernel_launch(void* const* d_in, const int* in_sizes, int n_in,
                              void* d_out, int out_size, void* d_ws, size_t ws_size,
                              hipStream_t stream) {
    (void)in_sizes; (void)n_in; (void)out_size; (void)ws_size;
    const float* x = (const float*)d_in[0];   // (200000, 256) f32
    const float* W = (const float*)d_in[1];   // (2, 256, 256) f32
    float* out = (float*)d_out;               // (1, 512) f32

    float* ws      = (float*)d_ws;
    float* cs_part = ws;                               // CS_BLOCKS*256
    float* temp1   = cs_part + CS_BLOCKS * EMB;        // 256
    float* h1      = temp1 + EMB;                      // 512
    float* h2      = h1 + 512;                         // 512
    float* att1    = h2 + 512;                         // 2*N
    float* part    = att1 + 2 * (size_t)N_NODES;       // PASS_BLOCKS*512
    float* temp2   = part + PASS_BLOCKS * 512;         // 512

    // temp1 = mean(x) (shared by both heads)
    colsum_partial<<<CS_BLOCKS, 256, 0, stream>>>(x, cs_part);
    reduce_colsum<<<1, 256, 0, stream>>>(cs_part, temp1);
    // h1[i] = tanh(temp1 @ W[i])
    proj_wmma<<<32, 32, 0, stream>>>(temp1, 0, W, h1);
    // iteration 1: att1, out1 partials
    pass_attn<0><<<PASS_BLOCKS, 256, 0, stream>>>(x, h1, nullptr, att1, part);
    // temp2[i] = out1[i]/N  (== mean(x*att1))
    reduce_part_scale<<<1, 256, 0, stream>>>(part, PASS_BLOCKS, temp2, 1.0f / (float)N_NODES);
    // h2[i] = tanh(temp2[i] @ W[i])
    proj_wmma<<<32, 32, 0, stream>>>(temp2, EMB, W, h2);
    // iteration 2: out2 partials -> d_out
    pass_attn<1><<<PASS_BLOCKS, 256, 0, stream>>>(x, h2, att1, nullptr, part);
    reduce_part_scale<<<1, 256, 0, stream>>>(part, PASS_BLOCKS, out, 1.0f);
}